// GCNEncoderNet_46866683134564
// MI455X (gfx1250) — compile-verified
//
#include <hip/hip_runtime.h>

typedef __attribute__((ext_vector_type(2))) float v2f;
typedef __attribute__((ext_vector_type(8))) float v8f;

#define F_IN   128
#define H_DIM  64
#define Z_DIM  32
#define G_GRAPHS 512

// ---------------------------------------------------------------------------
// Generic fp32 WMMA GEMM: C[M x Nc] = A[M x K] @ B[K x Nc] (+ bias per col)
// One wave (32 lanes) computes one 16x16 output tile via V_WMMA_F32_16X16X4_F32.
// Layouts per CDNA5 ISA 7.12.2:
//   A (16x4, 2 VGPRs): lanes 0-15 -> M=lane, K=k..k+1 ; lanes 16-31 -> K=k+2..k+3
//   B (4x16, 2 VGPRs): lanes 0-15 -> N=lane, K=k..k+1 ; lanes 16-31 -> K=k+2..k+3
//   C/D (8 VGPRs): VGPR r: lanes 0-15 -> M=r, lanes 16-31 -> M=r+8; N=lane%16
// ---------------------------------------------------------------------------
__global__ __launch_bounds__(32) void gemm_wmma_f32(
    const float* __restrict__ A, const float* __restrict__ B,
    const float* __restrict__ bias, float* __restrict__ C,
    int M, int K, int Nc)
{
  const int tileN = blockIdx.x;            // Nc / 16 tiles
  const int tileM = blockIdx.y;            // M  / 16 tiles
  const int lane  = threadIdx.x;           // 0..31, EXEC all ones (no guards)
  const int half  = lane >> 4;             // 0: K pair {k,k+1}, 1: {k+2,k+3}
  const int l     = lane & 15;

  const float* Arow = A + (long long)(tileM * 16 + l) * K;   // this lane's M row
  const float* Bcol = B + (tileN * 16 + l);                  // this lane's N column

  v8f c = {0.f, 0.f, 0.f, 0.f, 0.f, 0.f, 0.f, 0.f};

  for (int k = 0; k < K; k += 4) {
    const int ka = k + 2 * half;
    v2f a, b;
    a.x = Arow[ka];
    a.y = Arow[ka + 1];
    b.x = Bcol[(long long)ka * Nc];
    b.y = Bcol[(long long)(ka + 1) * Nc];
    // 8 args: (neg_a, A, neg_b, B, c_mod, C, reuse_a, reuse_b)
    c = __builtin_amdgcn_wmma_f32_16x16x4_f32(
        false, a, false, b, (short)0, c, false, false);
  }

  const int col = tileN * 16 + l;
  const float bv = bias ? bias[col] : 0.0f;
  float* Crow = C + (long long)(tileM * 16 + 8 * half) * Nc + col;
#pragma unroll
  for (int r = 0; r < 8; ++r) {
    Crow[(long long)r * Nc] = c[r] + bv;
  }
}

// ---------------------------------------------------------------------------
// Elementwise / scatter kernels
// ---------------------------------------------------------------------------
__global__ void fill_f32(float* __restrict__ p, float v, int n) {
  int t = blockIdx.x * blockDim.x + threadIdx.x;
  if (t < n) p[t] = v;
}

__global__ void deg_accum(const int* __restrict__ dst, float* __restrict__ deg, int E) {
  int t = blockIdx.x * blockDim.x + threadIdx.x;
  if (t < E) atomicAdd(&deg[dst[t]], 1.0f);
}

__global__ void deg_to_dinv(float* __restrict__ d, int n) {
  int t = blockIdx.x * blockDim.x + threadIdx.x;
  if (t < n) d[t] = rsqrtf(d[t] + 1.0f);
}

// agg[i][f] = h[i][f] * dinv[i]^2   (self-loop term, also initializes agg)
__global__ void selfloop_init(const float* __restrict__ h,
                              const float* __restrict__ dinv,
                              float* __restrict__ agg, int n) {
  int t = blockIdx.x * blockDim.x + threadIdx.x;
  if (t >= n * H_DIM) return;
  int i = t / H_DIM;
  float di = dinv[i];
  agg[t] = h[t] * (di * di);
}

// agg[dst] += h[src] * dinv[src]*dinv[dst]   (float4 per thread, 16 chunks/edge)
__global__ void edge_prop(const int* __restrict__ src, const int* __restrict__ dst,
                          const float* __restrict__ dinv,
                          const float* __restrict__ h, float* __restrict__ agg,
                          int E) {
  const int chunks = H_DIM / 4;
  long long t = (long long)blockIdx.x * blockDim.x + threadIdx.x;
  if (t >= (long long)E * chunks) return;
  int e = (int)(t / chunks);
  int c = (int)(t % chunks);
  int s = src[e], d = dst[e];
  float coef = dinv[s] * dinv[d];
  float4 v = ((const float4*)(h + (long long)s * H_DIM))[c];
  float* o = agg + (long long)d * H_DIM + 4 * c;
  atomicAdd(o + 0, v.x * coef);
  atomicAdd(o + 1, v.y * coef);
  atomicAdd(o + 2, v.z * coef);
  atomicAdd(o + 3, v.w * coef);
}

__global__ void bias_relu(const float* __restrict__ agg, const float* __restrict__ b,
                          float* __restrict__ out, int n) {
  int t = blockIdx.x * blockDim.x + threadIdx.x;
  if (t >= n * H_DIM) return;
  float v = agg[t] + b[t % H_DIM];
  out[t] = v > 0.0f ? v : 0.0f;
}

__global__ void pool_accum(const float* __restrict__ h, const int* __restrict__ batch,
                           float* __restrict__ sums, float* __restrict__ cnts, int n) {
  const int chunks = H_DIM / 4;
  int t = blockIdx.x * blockDim.x + threadIdx.x;
  if (t >= n * chunks) return;
  int i = t / chunks, c = t % chunks;
  int g = batch[i];
  float4 v = ((const float4*)(h + (long long)i * H_DIM))[c];
  float* o = sums + (long long)g * H_DIM + 4 * c;
  atomicAdd(o + 0, v.x);
  atomicAdd(o + 1, v.y);
  atomicAdd(o + 2, v.z);
  atomicAdd(o + 3, v.w);
  if (c == 0) atomicAdd(&cnts[g], 1.0f);
}

__global__ void pool_final(const float* __restrict__ sums, const float* __restrict__ cnts,
                           float* __restrict__ pooled) {
  int t = blockIdx.x * blockDim.x + threadIdx.x;
  if (t >= G_GRAPHS * H_DIM) return;
  pooled[t] = sums[t] / fmaxf(cnts[t / H_DIM], 1.0f);
}

// ---------------------------------------------------------------------------
// Host-side orchestration
// ---------------------------------------------------------------------------
static inline size_t align_up(size_t x, size_t a) { return (x + a - 1) & ~(a - 1); }

extern "C" void kernel_launch(void* const* d_in, const int* in_sizes, int n_in,
                              void* d_out, int out_size, void* d_ws, size_t ws_size,
                              hipStream_t stream) {
  const float* x   = (const float*)d_in[0];
  const int*   ei  = (const int*)d_in[1];
  const int*   bat = (const int*)d_in[2];
  const float* W1  = (const float*)d_in[3];
  const float* b1  = (const float*)d_in[4];
  const float* W2  = (const float*)d_in[5];
  const float* b2  = (const float*)d_in[6];
  const float* Wmu = (const float*)d_in[7];
  const float* bmu = (const float*)d_in[8];
  const float* Wls = (const float*)d_in[9];
  const float* bls = (const float*)d_in[10];
  float* out = (float*)d_out;

  const int N = in_sizes[0] / F_IN;     // 50000
  const int E = in_sizes[1] / 2;        // 800000
  const int* src = ei;
  const int* dst = ei + E;

  // workspace layout
  char* ws = (char*)d_ws;
  size_t off = 0;
  float* dinv = (float*)(ws + off); off = align_up(off + (size_t)N * 4, 256);
  float* bufA = (float*)(ws + off); off = align_up(off + (size_t)N * H_DIM * 4, 256);
  float* bufB = (float*)(ws + off); off = align_up(off + (size_t)N * H_DIM * 4, 256);
  float* bufC = (float*)(ws + off); off = align_up(off + (size_t)N * H_DIM * 4, 256);
  float* sums = (float*)(ws + off); off = align_up(off + (size_t)G_GRAPHS * H_DIM * 4, 256);
  float* cnts = (float*)(ws + off); off = align_up(off + (size_t)G_GRAPHS * 4, 256);
  float* pooled = (float*)(ws + off); off = align_up(off + (size_t)G_GRAPHS * H_DIM * 4, 256);
  (void)ws_size; (void)n_in; (void)out_size;

  const int BLK = 256;
  const int NH = N * H_DIM;
  const long long EC = (long long)E * (H_DIM / 4);
  const int NC = N * (H_DIM / 4);

  // degree -> dinv (deterministic: re-zeroed every call)
  fill_f32<<<(N + BLK - 1) / BLK, BLK, 0, stream>>>(dinv, 0.0f, N);
  deg_accum<<<(E + BLK - 1) / BLK, BLK, 0, stream>>>(dst, dinv, E);
  deg_to_dinv<<<(N + BLK - 1) / BLK, BLK, 0, stream>>>(dinv, N);

  // ---- layer 1: h1 = x @ W1 ; agg = prop(h1) ; bufA = relu(agg + b1)
  gemm_wmma_f32<<<dim3(H_DIM / 16, N / 16), 32, 0, stream>>>(
      x, W1, nullptr, bufA, N, F_IN, H_DIM);
  selfloop_init<<<(NH + BLK - 1) / BLK, BLK, 0, stream>>>(bufA, dinv, bufB, N);
  edge_prop<<<(unsigned)((EC + BLK - 1) / BLK), BLK, 0, stream>>>(
      src, dst, dinv, bufA, bufB, E);
  bias_relu<<<(NH + BLK - 1) / BLK, BLK, 0, stream>>>(bufB, b1, bufA, N);

  // ---- layer 2: h2 = bufA @ W2 ; agg = prop(h2) ; bufA = relu(agg + b2)
  gemm_wmma_f32<<<dim3(H_DIM / 16, N / 16), 32, 0, stream>>>(
      bufA, W2, nullptr, bufB, N, H_DIM, H_DIM);
  selfloop_init<<<(NH + BLK - 1) / BLK, BLK, 0, stream>>>(bufB, dinv, bufC, N);
  edge_prop<<<(unsigned)((EC + BLK - 1) / BLK), BLK, 0, stream>>>(
      src, dst, dinv, bufB, bufC, E);
  bias_relu<<<(NH + BLK - 1) / BLK, BLK, 0, stream>>>(bufC, b2, bufA, N);

  // ---- global mean pool
  fill_f32<<<(G_GRAPHS * H_DIM + BLK - 1) / BLK, BLK, 0, stream>>>(sums, 0.0f, G_GRAPHS * H_DIM);
  fill_f32<<<(G_GRAPHS + BLK - 1) / BLK, BLK, 0, stream>>>(cnts, 0.0f, G_GRAPHS);
  pool_accum<<<(NC + BLK - 1) / BLK, BLK, 0, stream>>>(bufA, bat, sums, cnts, N);
  pool_final<<<(G_GRAPHS * H_DIM + BLK - 1) / BLK, BLK, 0, stream>>>(sums, cnts, pooled);

  // ---- heads: mu then logsigma, concatenated flat in d_out
  gemm_wmma_f32<<<dim3(Z_DIM / 16, G_GRAPHS / 16), 32, 0, stream>>>(
      pooled, Wmu, bmu, out, G_GRAPHS, H_DIM, Z_DIM);
  gemm_wmma_f32<<<dim3(Z_DIM / 16, G_GRAPHS / 16), 32, 0, stream>>>(
      pooled, Wls, bls, out + (size_t)G_GRAPHS * Z_DIM, G_GRAPHS, H_DIM, Z_DIM);
}